// Yolov8FullModel_10977936408639
// MI455X (gfx1250) — compile-verified
//
#include <hip/hip_runtime.h>
#include <stdint.h>

#define N_ANCH   134400
#define N_CLS    80
#define KTOP     4096
#define MAXOUT   300
#define CONF_T   0.25f
#define CELL     640.0f

typedef __attribute__((ext_vector_type(2))) float v2f;
typedef __attribute__((ext_vector_type(8))) float v8f;
typedef __attribute__((ext_vector_type(4))) unsigned int u32x4;
typedef __attribute__((ext_vector_type(8))) int i32x8;
typedef __attribute__((ext_vector_type(4))) int i32x4;

// ---------------------------------------------------------------- K1: decode
__global__ void k_decode(const float* __restrict__ p, unsigned int* __restrict__ keys,
                         float* __restrict__ boxes, int* __restrict__ cls) {
    int a = blockIdx.x * blockDim.x + threadIdx.x;
    if (a >= N_ANCH) return;
    float best = -1.0f; int bc = 0;
    #pragma unroll 8
    for (int c = 0; c < N_CLS; ++c) {
        float v = p[(size_t)(4 + c) * N_ANCH + a];
        if (v > best) { best = v; bc = c; }
    }
    float xc = p[(size_t)0 * N_ANCH + a];
    float yc = p[(size_t)1 * N_ANCH + a];
    float w  = p[(size_t)2 * N_ANCH + a];
    float h  = p[(size_t)3 * N_ANCH + a];
    float hw = floorf(w * 0.5f), hh = floorf(h * 0.5f);
    boxes[a * 4 + 0] = xc - hw;
    boxes[a * 4 + 1] = yc - hh;
    boxes[a * 4 + 2] = xc + hw;
    boxes[a * 4 + 3] = yc + hh;
    keys[a] = (best >= CONF_T) ? __float_as_uint(best) : 0u;
    cls[a] = bc;
}

// ------------------------------------------- K2a: radix-select 4096th value
// sel[0]=pivot key, sel[1]=#strictly greater, sel[2..3]=compaction counters
__global__ void k_select(const unsigned int* __restrict__ keys,
                         unsigned int* __restrict__ sel,
                         unsigned long long* __restrict__ pairs) {
    __shared__ unsigned int hist[256];
    __shared__ unsigned int s_prefix, s_remaining, s_kstrict;
    int tid = threadIdx.x;
    for (int i = tid; i < KTOP; i += blockDim.x) pairs[i] = 0ull;  // sentinel
    if (tid == 0) { s_prefix = 0; s_remaining = KTOP; s_kstrict = 0; sel[2] = 0; sel[3] = 0; }
    for (int pass = 3; pass >= 0; --pass) {
        int shift = pass * 8;
        if (tid < 256) hist[tid] = 0;
        __syncthreads();
        unsigned int prefix = s_prefix;
        unsigned int himask = (pass == 3) ? 0u : (0xFFFFFFFFu << (shift + 8));
        for (int i = tid; i < N_ANCH; i += blockDim.x) {
            unsigned int k = keys[i];
            if ((k & himask) == (prefix & himask))
                atomicAdd(&hist[(k >> shift) & 255], 1u);
        }
        __syncthreads();
        if (tid == 0) {
            unsigned int rem = s_remaining, cum = 0;
            int b = 255;
            while (b > 0 && cum + hist[b] < rem) { cum += hist[b]; --b; }
            s_prefix = prefix | ((unsigned int)b << shift);
            s_remaining = rem - cum;
            s_kstrict += cum;
            if (pass == 0) { sel[0] = s_prefix; sel[1] = s_kstrict; }
        }
        __syncthreads();
    }
}

// ------------------------------------------------------ K2b: compact top-K
__global__ void k_compact(const unsigned int* __restrict__ keys,
                          unsigned int* __restrict__ sel,
                          unsigned long long* __restrict__ pairs) {
    int i = blockIdx.x * blockDim.x + threadIdx.x;
    if (i >= N_ANCH) return;
    unsigned int k = keys[i];
    unsigned int pivot = sel[0], kstrict = sel[1];
    unsigned long long v = ((unsigned long long)k << 32) | (unsigned int)(~i);
    if (k > pivot) {
        unsigned int pos = atomicAdd(&sel[2], 1u);
        if (pos < KTOP) pairs[pos] = v;
    } else if (k == pivot) {
        unsigned int p = atomicAdd(&sel[3], 1u);
        unsigned int pos = kstrict + p;
        if (kstrict + p < KTOP && p < KTOP - kstrict) pairs[pos] = v;
    }
}

// --------------- K2c: TDM-stage pairs into LDS, bitonic sort, gather cands
__global__ void k_sort(const unsigned long long* __restrict__ pairs_g,
                       const float* __restrict__ boxes, const int* __restrict__ cls,
                       unsigned int* __restrict__ cand_key, float* __restrict__ cand_box,
                       float* __restrict__ cand_area, int* __restrict__ cand_cls) {
    __shared__ unsigned long long s[KTOP];   // 32 KB
    int tid = threadIdx.x;

    // ---- Tensor Data Mover: async DMA 4096 x 8B tile (global -> LDS) ----
    if (tid < 32) {
        unsigned long long ga = (unsigned long long)(uintptr_t)pairs_g;
        unsigned int lds = (unsigned int)(uintptr_t)&s[0];
        u32x4 g0 = { 1u,                               // count=1 valid descriptor
                     lds,                              // lds_addr
                     (unsigned int)(ga & 0xFFFFFFFFu), // global_addr[31:0]
                     (unsigned int)(((ga >> 32) & 0x1FFFFFFu) | (2u << 30)) }; // [56:32], type=2
        // group1: data_size=8B, tensor_dim0=4096, tensor_dim1=1, tile_dim0=4096,
        //         tile_dim1=1, tensor_dim0_stride=4096
        i32x8 g1 = { (int)0x00030000,
                     (int)((KTOP & 0xFFFF) << 16),
                     (int)(1 << 16),
                     (int)((KTOP & 0xFFFF) << 16),
                     1,
                     KTOP,
                     0, 0 };
        i32x4 g2 = { 0, 0, 0, 0 };
        i32x4 g3 = { 0, 0, 0, 0 };
#if __clang_major__ >= 23
        i32x8 g4 = { 0, 0, 0, 0, 0, 0, 0, 0 };
        __builtin_amdgcn_tensor_load_to_lds(g0, g1, g2, g3, g4, 0);
#else
        __builtin_amdgcn_tensor_load_to_lds(g0, g1, g2, g3, 0);
#endif
        __builtin_amdgcn_s_wait_tensorcnt(0);
    }
    __syncthreads();

    // ---- bitonic sort, descending by composite (score desc, idx asc) ----
    for (unsigned int k = 2; k <= KTOP; k <<= 1) {
        for (unsigned int j = k >> 1; j > 0; j >>= 1) {
            __syncthreads();
            for (unsigned int t = tid; t < KTOP / 2; t += blockDim.x) {
                unsigned int i = ((t & ~(j - 1)) << 1) | (t & (j - 1));
                unsigned int ixj = i | j;
                bool up = ((i & k) == 0);
                unsigned long long a = s[i], b = s[ixj];
                if ((a < b) == up) { s[i] = b; s[ixj] = a; }
            }
        }
    }
    __syncthreads();

    // ---- gather candidate boxes / classes / areas in sorted order ----
    for (int i = tid; i < KTOP; i += blockDim.x) {
        unsigned long long v = s[i];
        unsigned int key = (unsigned int)(v >> 32);
        unsigned int idx = ~(unsigned int)(v & 0xFFFFFFFFu);
        cand_key[i] = key;
        float x0 = 0.f, y0 = 0.f, x1 = 0.f, y1 = 0.f; int c = 0;
        if (key != 0u && idx < (unsigned int)N_ANCH) {
            x0 = boxes[idx * 4 + 0]; y0 = boxes[idx * 4 + 1];
            x1 = boxes[idx * 4 + 2]; y1 = boxes[idx * 4 + 3];
            c = cls[idx];
        }
        cand_box[i * 4 + 0] = x0; cand_box[i * 4 + 1] = y0;
        cand_box[i * 4 + 2] = x1; cand_box[i * 4 + 3] = y1;
        cand_area[i] = (x1 - x0) * (y1 - y0);
        cand_cls[i] = c;
    }
}

// ------------- K3a: suppression bitmask via WMMA row-area broadcast + VALU IoU ------
// grid (32, 256), block 256 = 8 waves; block tile: 16 rows x 128 cols.
__global__ void k_iou_mask(const float* __restrict__ cand_box,
                           const float* __restrict__ cand_area,
                           unsigned short* __restrict__ mask) {
    __shared__ float rb[16 * 4];
    __shared__ float ra[16];
    __shared__ float cb[128 * 4];
    __shared__ float ca[128];
    int bi0 = blockIdx.y * 16;
    int bj0 = blockIdx.x * 128;
    int tid = threadIdx.x;
    if (tid < 64) rb[tid] = cand_box[bi0 * 4 + tid];
    else if (tid < 80) ra[tid - 64] = cand_area[bi0 + (tid - 64)];
    if (tid < 128) ca[tid] = cand_area[bj0 + tid];
    for (int t = tid; t < 512; t += 256) cb[t] = cand_box[bj0 * 4 + t];
    __syncthreads();

    int wave = tid >> 5, lane = tid & 31;
    int cj0 = wave * 16;

    // A = [area_i, 1, 0, 0] (16x4), B = [1 ; 0 ; area_j ; 0] (4x16)  ->  D[i][j] = area_i
    v2f A, B;
    A.x = (lane < 16) ? ra[lane] : 0.0f;
    A.y = (lane < 16) ? 1.0f : 0.0f;
    B.x = (lane < 16) ? 1.0f : ca[cj0 + (lane - 16)];
    B.y = 0.0f;
    v8f C = {0.f, 0.f, 0.f, 0.f, 0.f, 0.f, 0.f, 0.f};
    v8f S = __builtin_amdgcn_wmma_f32_16x16x4_f32(false, A, false, B, (short)0, C,
                                                  false, false);

    int ibase = (lane >> 4) * 8;          // rows: lanes 0-15 -> M=r, 16-31 -> M=r+8
    int n = lane & 15;                    // column within tile
    int gj = bj0 + cj0 + n;
    float bx0 = cb[(cj0 + n) * 4 + 0], by0 = cb[(cj0 + n) * 4 + 1];
    float bx1 = cb[(cj0 + n) * 4 + 2], by1 = cb[(cj0 + n) * 4 + 3];
    int chunk = (bj0 + cj0) >> 4;

    #pragma unroll
    for (int r = 0; r < 8; ++r) {
        int il = ibase + r;
        int gi = bi0 + il;
        float ax0 = rb[il * 4 + 0], ay0 = rb[il * 4 + 1];
        float ax1 = rb[il * 4 + 2], ay1 = rb[il * 4 + 3];
        float ltx = fmaxf(ax0, bx0), lty = fmaxf(ay0, by0);
        float rbx = fminf(ax1, bx1), rby = fminf(ay1, by1);
        float w = fmaxf(rbx - ltx, 0.0f), h = fmaxf(rby - lty, 0.0f);
        float inter = w * h;
        float uni = S[r] - inter;         // area_i - inter (WMMA gives area_i only)
        // iou > 0.5  <=>  inter > 0.5*(union + 1e-9)
        bool pred = (gj > gi) && (inter > 0.5f * (uni + 1e-9f));
        unsigned long long bal = __ballot((int)pred);
        if (lane == 0) {
            mask[(bi0 + r) * 256 + chunk]     = (unsigned short)(bal & 0xFFFFull);
            mask[(bi0 + r + 8) * 256 + chunk] = (unsigned short)((bal >> 16) & 0xFFFFull);
        }
    }
}

// ---------------- K3b: greedy NMS reduce, single wave32, mask in VGPRs ----
__global__ void k_nms_reduce(const unsigned int* __restrict__ cand_key,
                             const unsigned short* __restrict__ mask16,
                             unsigned int* __restrict__ keepw) {
    int lane = threadIdx.x;                       // 32 threads = 1 wave
    const unsigned int* mask32 = (const unsigned int*)mask16;  // row stride 128 words
    unsigned int remv0 = 0, remv1 = 0, remv2 = 0, remv3 = 0;
    // invalid candidates start removed
    for (int t = 0; t < 4; ++t) {
        unsigned int m = 0;
        for (int b = 0; b < 32; ++b) {
            int j = lane * 128 + t * 32 + b;
            if (cand_key[j] == 0u) m |= (1u << b);
        }
        if (t == 0) remv0 = m; else if (t == 1) remv1 = m;
        else if (t == 2) remv2 = m; else remv3 = m;
    }
    for (int i = 0; i < KTOP; ++i) {
        int w = i >> 5, owner = w >> 2, t = w & 3;
        int keep = 0;
        if (lane == owner) {
            unsigned int rv = (t == 0) ? remv0 : (t == 1) ? remv1 : (t == 2) ? remv2 : remv3;
            keep = (int)(((rv >> (i & 31)) & 1u) ^ 1u);
        }
        keep = __shfl(keep, owner, 32);
        if (keep) {
            const unsigned int* row = mask32 + (size_t)i * 128 + lane * 4;
            remv0 |= row[0]; remv1 |= row[1]; remv2 |= row[2]; remv3 |= row[3];
        }
    }
    keepw[lane * 4 + 0] = ~remv0;
    keepw[lane * 4 + 1] = ~remv1;
    keepw[lane * 4 + 2] = ~remv2;
    keepw[lane * 4 + 3] = ~remv3;
}

// -------------------- K4: scan keeps, emit top-300, tie factor, count -----
__global__ void k_finalize(const unsigned int* __restrict__ keepw,
                           const unsigned int* __restrict__ cand_key,
                           const float* __restrict__ cand_box,
                           const int* __restrict__ cand_cls,
                           float* __restrict__ out) {
    __shared__ unsigned int psum[1024];
    __shared__ float s299;
    __shared__ int tieflag;
    __shared__ unsigned int cnt_total;
    int tid = threadIdx.x;
    unsigned int bits = (keepw[tid >> 3] >> ((tid & 7) * 4)) & 0xFu; // cands 4*tid..+3
    unsigned int cnt = __popc(bits);
    psum[tid] = cnt;
    __syncthreads();
    for (int off = 1; off < 1024; off <<= 1) {
        unsigned int v = (tid >= off) ? psum[tid - off] : 0u;
        __syncthreads();
        psum[tid] += v;
        __syncthreads();
    }
    unsigned int rank = psum[tid] - cnt;
    for (int q = 0; q < 4; ++q) {
        if (bits & (1u << q)) {
            if (rank < MAXOUT) {
                int j = 4 * tid + q;
                out[1200 + rank] = __uint_as_float(cand_key[j]);
                out[1500 + rank] = (float)cand_cls[j];
                out[rank * 4 + 0] = cand_box[j * 4 + 0] * (1.0f / CELL);
                out[rank * 4 + 1] = cand_box[j * 4 + 1] * (1.0f / CELL);
                out[rank * 4 + 2] = cand_box[j * 4 + 2] * (1.0f / CELL);
                out[rank * 4 + 3] = cand_box[j * 4 + 3] * (1.0f / CELL);
            }
            rank++;
        }
    }
    unsigned int total = psum[1023];
    if (total < MAXOUT) {
        for (unsigned int r = total + tid; r < MAXOUT; r += 1024) {
            out[1200 + r] = 0.0f; out[1500 + r] = 0.0f;
            out[r * 4 + 0] = 0.0f; out[r * 4 + 1] = 0.0f;
            out[r * 4 + 2] = 0.0f; out[r * 4 + 3] = 0.0f;
        }
    }
    __syncthreads();
    if (tid == 0) {
        float a = out[1200 + 298], b = out[1200 + 299];
        s299 = b;
        tieflag = (fabsf(b - a) < 1e-6f) ? 1 : 0;
        cnt_total = 0;
    }
    __syncthreads();
    if (tieflag) {
        for (int r = tid; r < MAXOUT; r += 1024) {
            float v = out[1200 + r];
            out[1200 + r] = (v - s299 > 1e-5f) ? v : 0.0f;
        }
    }
    __syncthreads();
    unsigned int c = 0;
    for (int r = tid; r < MAXOUT; r += 1024)
        if (out[1200 + r] > 1e-5f) c++;
    if (c) atomicAdd(&cnt_total, c);
    __syncthreads();
    if (tid == 0) out[1800] = (float)cnt_total;
}

// --------------------------------------------------------------------------
extern "C" void kernel_launch(void* const* d_in, const int* in_sizes, int n_in,
                              void* d_out, int out_size, void* d_ws, size_t ws_size,
                              hipStream_t stream) {
    const float* preds = (const float*)d_in[0];   // (1, 84, 134400) f32
    float* out = (float*)d_out;                   // 300*4 + 300 + 300 + 1 = 1801 f32

    char* base = (char*)d_ws;
    size_t o = 0;
    auto alloc = [&](size_t bytes) { size_t r = o; o += (bytes + 255) & ~(size_t)255; return r; };
    unsigned int* keys        = (unsigned int*)(base + alloc((size_t)N_ANCH * 4));
    float* boxes              = (float*)(base + alloc((size_t)N_ANCH * 16));
    int* cls                  = (int*)(base + alloc((size_t)N_ANCH * 4));
    unsigned long long* pairs = (unsigned long long*)(base + alloc((size_t)KTOP * 8));
    unsigned int* sel         = (unsigned int*)(base + alloc(64));
    unsigned int* cand_key    = (unsigned int*)(base + alloc((size_t)KTOP * 4));
    float* cand_box           = (float*)(base + alloc((size_t)KTOP * 16));
    float* cand_area          = (float*)(base + alloc((size_t)KTOP * 4));
    int* cand_cls             = (int*)(base + alloc((size_t)KTOP * 4));
    unsigned short* mask      = (unsigned short*)(base + alloc((size_t)KTOP * 256 * 2));
    unsigned int* keepw       = (unsigned int*)(base + alloc(128 * 4));
    (void)ws_size; (void)in_sizes; (void)n_in; (void)out_size;

    k_decode<<<(N_ANCH + 255) / 256, 256, 0, stream>>>(preds, keys, boxes, cls);
    k_select<<<1, 1024, 0, stream>>>(keys, sel, pairs);
    k_compact<<<(N_ANCH + 255) / 256, 256, 0, stream>>>(keys, sel, pairs);
    k_sort<<<1, 1024, 0, stream>>>(pairs, boxes, cls, cand_key, cand_box, cand_area, cand_cls);
    k_iou_mask<<<dim3(32, 256), 256, 0, stream>>>(cand_box, cand_area, mask);
    k_nms_reduce<<<1, 32, 0, stream>>>(cand_key, mask, keepw);
    k_finalize<<<1, 1024, 0, stream>>>(keepw, cand_key, cand_box, cand_cls, out);
}